// ClusterMerging_71511205479197
// MI455X (gfx1250) — compile-verified
//
#include <hip/hip_runtime.h>
#include <hip/hip_bf16.h>
#include <math.h>

typedef __attribute__((ext_vector_type(2))) float v2f;
typedef __attribute__((ext_vector_type(8))) float v8f;

#define BB    4
#define HH    112
#define WW    112
#define NN    (HH*WW)          /* 12544 tokens */
#define NBHD  48
#define DIM   256
#define OUTD  512
#define INNER 4
#define TABLE 4096
#define KEEP  (NN/4)           /* 3136 */
#define RESV  ((HH/4)*(WW/4))  /* 784  */
#define SAMP  (KEEP-RESV)      /* 2352 */
#define XDIM  (INNER*DIM)      /* 1024 */

// ---------------------------------------------------------------------------
// Kernel 1: weight_net on the PE table: wt = gelu(LN(pre_table @ w1 + b1))
// ---------------------------------------------------------------------------
__global__ void wt_kernel(const float* __restrict__ pre, const float* __restrict__ w1,
                          const float* __restrict__ b1, const float* __restrict__ g,
                          const float* __restrict__ bta, float* __restrict__ wt) {
    int t = blockIdx.x * blockDim.x + threadIdx.x;
    if (t >= TABLE) return;
    float p[5];
#pragma unroll
    for (int j = 0; j < 5; j++) p[j] = pre[t * 5 + j];
    float h[INNER];
#pragma unroll
    for (int i = 0; i < INNER; i++) {
        float s = b1[i];
#pragma unroll
        for (int j = 0; j < 5; j++) s += p[j] * w1[j * INNER + i];
        h[i] = s;
    }
    float m = 0.25f * (h[0] + h[1] + h[2] + h[3]);
    float v = 0.f;
#pragma unroll
    for (int i = 0; i < INNER; i++) { float d = h[i] - m; v += d * d; }
    v *= 0.25f;
    float r = rsqrtf(v + 1e-5f);
#pragma unroll
    for (int i = 0; i < INNER; i++) {
        float x = (h[i] - m) * r * g[i] + bta[i];
        wt[t * INNER + i] = 0.5f * x * (1.0f + erff(x * 0.70710678118654752f));
    }
}

// ---------------------------------------------------------------------------
// Kernel 2: order-preserving u64 sort keys for stable descending top-k
// key = ord(score) << 32 | (~index)  => key_j > key_i  <=>  j ranks before i
// ---------------------------------------------------------------------------
__global__ void key_kernel(const float* __restrict__ pos, const float* __restrict__ lp,
                           unsigned long long* __restrict__ keys) {
    int t = blockIdx.x * blockDim.x + threadIdx.x;   // b*N + n
    if (t >= BB * NN) return;
    int n = t % NN;
    int pi = (int)pos[(size_t)t * 2 + 0];
    int pj = (int)pos[(size_t)t * 2 + 1];
    float fp = lp[t] * 4.0f;                              // ALPHA * learned_prob
    if (((pi & 1) == 0) && ((pj & 1) == 0)) fp += 1.0f;   // stride-2 grid bonus
    if (((pi & 3) == 0) && ((pj & 3) == 0)) fp -= 100.0f; // reserve penalty
    unsigned u = __float_as_uint(fp);
    u = (u & 0x80000000u) ? ~u : (u | 0x80000000u);       // monotonic float->uint
    keys[t] = ((unsigned long long)u << 32) |
              (unsigned long long)(0xFFFFFFFFu - (unsigned)n);
}

// ---------------------------------------------------------------------------
// Kernel 3: exact rank-select. rank_i = #{j : key_j > key_i}; rank < SAMP
// scatters to idx[rank] (== jax top_k order). Reserve tokens fill the tail
// in increasing-index order via closed form.
// ---------------------------------------------------------------------------
__global__ void select_kernel(const unsigned long long* __restrict__ keys,
                              const float* __restrict__ pos, int* __restrict__ idx) {
    __shared__ unsigned long long kt[256];
    int b = blockIdx.x / (NN / 256);
    int i = (blockIdx.x % (NN / 256)) * 256 + threadIdx.x;
    const unsigned long long* kb = keys + (size_t)b * NN;
    unsigned long long my = kb[i];
    int rank = 0;
    for (int jt = 0; jt < NN; jt += 256) {
        kt[threadIdx.x] = kb[jt + threadIdx.x];
        __syncthreads();
#pragma unroll 8
        for (int jj = 0; jj < 256; jj++) rank += (kt[jj] > my) ? 1 : 0;
        __syncthreads();
    }
    if (rank < SAMP) idx[b * KEEP + rank] = i;
    int pi = (int)pos[((size_t)b * NN + i) * 2 + 0];
    int pj = (int)pos[((size_t)b * NN + i) * 2 + 1];
    if (((pi & 3) == 0) && ((pj & 3) == 0)) {
        int r = (pi >> 2) * (WW / 4) + (pj >> 2);
        idx[b * KEEP + SAMP + r] = i;
    }
}

// ---------------------------------------------------------------------------
// Kernel 4: per kept token: stage 48x4 edge weights in LDS, coalesced gather
// of 48 feature rows, 4-way weighted accumulation, block LayerNorm(1024),
// write normalized X to scratch and pos_k to d_out.
// ---------------------------------------------------------------------------
__global__ void gather_kernel(const int* __restrict__ idx, const int* __restrict__ mem,
                              const int* __restrict__ pe, const float* __restrict__ cm,
                              const float* __restrict__ lp, const float* __restrict__ pos,
                              const float* __restrict__ feat, const float* __restrict__ wt,
                              const float* __restrict__ ng, const float* __restrict__ nb,
                              float* __restrict__ posk, float* __restrict__ Xn) {
    __shared__ int   memL[NBHD];
    __shared__ float wL[NBHD * INNER];
    __shared__ float red[256];

    int t = blockIdx.x;                 // b*KEEP + nk
    int b = t / KEEP;
    int i = idx[t];
    size_t base = (size_t)b * NN + (size_t)i;

    if (threadIdx.x < NBHD) {
        int k  = threadIdx.x;
        int mk = mem[base * NBHD + k];
        int pk = pe[base * NBHD + k];
        float w = lp[(size_t)b * NN + mk] * cm[base * NBHD + k];
        memL[k] = mk;
#pragma unroll
        for (int m = 0; m < INNER; m++) wL[k * INNER + m] = wt[pk * INNER + m] * w;
    }
    if (threadIdx.x < 2)
        posk[(size_t)t * 2 + threadIdx.x] = pos[base * 2 + threadIdx.x];
    __syncthreads();

    int c = threadIdx.x;                // channel 0..255
    float a0 = 0.f, a1 = 0.f, a2 = 0.f, a3 = 0.f;
    const float* fb = feat + (size_t)b * NN * DIM;
    for (int k = 0; k < NBHD; k++) {
        const float* fr = fb + (size_t)memL[k] * DIM;
        if (k + 1 < NBHD)
            __builtin_prefetch(fb + (size_t)memL[k + 1] * DIM + c, 0, 0); // global_prefetch_b8
        float v = fr[c];
        a0 += wL[k * INNER + 0] * v;
        a1 += wL[k * INNER + 1] * v;
        a2 += wL[k * INNER + 2] * v;
        a3 += wL[k * INNER + 3] * v;
    }

    // block reduction: mean over 1024
    red[c] = a0 + a1 + a2 + a3;
    __syncthreads();
    for (int st = 128; st > 0; st >>= 1) {
        if (c < st) red[c] += red[c + st];
        __syncthreads();
    }
    float mean = red[0] * (1.0f / (float)XDIM);
    __syncthreads();
    // block reduction: E[x^2]
    red[c] = a0 * a0 + a1 * a1 + a2 * a2 + a3 * a3;
    __syncthreads();
    for (int st = 128; st > 0; st >>= 1) {
        if (c < st) red[c] += red[c + st];
        __syncthreads();
    }
    float var  = red[0] * (1.0f / (float)XDIM) - mean * mean;
    float rstd = rsqrtf(var + 1e-5f);

    float acc[INNER] = {a0, a1, a2, a3};
    float* xr = Xn + (size_t)t * XDIM;
#pragma unroll
    for (int m = 0; m < INNER; m++) {
        int col = m * DIM + c;
        xr[col] = (acc[m] - mean) * rstd * ng[col] + nb[col];
    }
}

// ---------------------------------------------------------------------------
// Kernel 5: Xn (12544 x 1024) @ lin_w (1024 x 512) + lin_b, f32 WMMA.
// One wave per 16x128 output tile; 8 x v8f accumulators; K in steps of 4.
// ---------------------------------------------------------------------------
__global__ void gemm_kernel(const float* __restrict__ Xn, const float* __restrict__ Wl,
                            const float* __restrict__ bias, float* __restrict__ out) {
    int wave = threadIdx.x >> 5;         // 4 waves / block
    int lane = threadIdx.x & 31;
    int m16  = blockIdx.x * 16;          // M tile
    int n0w  = wave * 128;               // N range for this wave
    int mrow = lane & 15;
    int kk   = (lane >> 4) * 2;          // A/B K sub-index per ISA layout

    v8f acc[8] = {};
    const float* xrow = Xn + (size_t)(m16 + mrow) * XDIM;

    for (int k0 = 0; k0 < XDIM; k0 += 4) {
        v2f a;
        a.x = xrow[k0 + kk];
        a.y = xrow[k0 + kk + 1];
        const float* w0 = Wl + (size_t)(k0 + kk) * OUTD;
        const float* w1 = Wl + (size_t)(k0 + kk + 1) * OUTD;
#pragma unroll
        for (int j = 0; j < 8; j++) {
            int n = n0w + j * 16 + mrow;
            v2f bf;
            bf.x = w0[n];
            bf.y = w1[n];
            acc[j] = __builtin_amdgcn_wmma_f32_16x16x4_f32(
                false, a, false, bf, (short)0, acc[j], false, false);
        }
    }

    int rbase = m16 + ((lane >> 4) * 8); // D layout: lanes 16-31 hold M=r+8
#pragma unroll
    for (int j = 0; j < 8; j++) {
        int ncol = n0w + j * 16 + (lane & 15);
        float bb = bias[ncol];
#pragma unroll
        for (int r = 0; r < 8; r++)
            out[(size_t)(rbase + r) * OUTD + ncol] = acc[j][r] + bb;
    }
}

// ---------------------------------------------------------------------------
extern "C" void kernel_launch(void* const* d_in, const int* in_sizes, int n_in,
                              void* d_out, int out_size, void* d_ws, size_t ws_size,
                              hipStream_t stream) {
    const float* pos  = (const float*)d_in[0];
    const float* feat = (const float*)d_in[1];
    const int*   mem  = (const int*)d_in[2];
    const float* cm   = (const float*)d_in[3];
    const float* lp   = (const float*)d_in[4];
    // d_in[5] = stride (2), d_in[7] = reserve_num (784): fixed by problem shape
    const int*   pe   = (const int*)d_in[6];
    const float* pre  = (const float*)d_in[8];
    const float* w1   = (const float*)d_in[9];
    const float* b1   = (const float*)d_in[10];
    const float* ln1g = (const float*)d_in[11];
    const float* ln1b = (const float*)d_in[12];
    const float* ng   = (const float*)d_in[13];
    const float* nb   = (const float*)d_in[14];
    const float* lw   = (const float*)d_in[15];
    const float* lb   = (const float*)d_in[16];

    char* ws = (char*)d_ws;
    float*              wt   = (float*)ws;                               // 65536 B
    unsigned long long* keys = (unsigned long long*)(ws + 65536);        // 401408 B
    int*                idx  = (int*)(ws + 65536 + 401408);              // 50176 B
    float*              Xn   = (float*)(ws + 65536 + 401408 + 50176);    // 51.4 MB

    float* posk = (float*)d_out;
    float* out2 = (float*)d_out + (size_t)BB * KEEP * 2;

    wt_kernel<<<TABLE / 256, 256, 0, stream>>>(pre, w1, b1, ln1g, ln1b, wt);
    key_kernel<<<(BB * NN) / 256, 256, 0, stream>>>(pos, lp, keys);
    select_kernel<<<BB * (NN / 256), 256, 0, stream>>>(keys, pos, idx);
    gather_kernel<<<BB * KEEP, 256, 0, stream>>>(idx, mem, pe, cm, lp, pos, feat,
                                                 wt, ng, nb, posk, Xn);
    gemm_kernel<<<(BB * KEEP) / 16, 128, 0, stream>>>(Xn, lw, lb, out2);
}